// GCN_76407468195986
// MI455X (gfx1250) — compile-verified
//
#include <hip/hip_runtime.h>
#include <hip/hip_bf16.h>

#define CDIM   128
#define BGRAPH 64
#define EDGES  2097152

typedef float v2f __attribute__((ext_vector_type(2)));
typedef float v8f __attribute__((ext_vector_type(8)));

// ---------------------------------------------------------------------------
// GEMM: Out[M,128] = A[M,128] @ W[128,128]  (row-major, W[k][o])
// WMMA f32 16x16x4. Block = 256 threads (8 waves), block tile = 64 x 128.
// Wave w: rows (w&3)*16..+16, cols (w>>2)*64..+64  -> 4 wmma tiles along N.
// A staged row-major (stride 132), W staged TRANSPOSED per K-chunk
// (Bst[col][k], stride 36) so every fragment is one aligned ds_load_b64.
// ---------------------------------------------------------------------------
__launch_bounds__(256)
__global__ void gcn_gemm_wmma(const float* __restrict__ A,
                              const float* __restrict__ W,
                              float* __restrict__ Out, int M)
{
    __shared__ float As[64 * 132];    // A tile, padded stride 132
    __shared__ float Bst[128 * 36];   // W K-chunk, transposed, padded stride 36

    const int tid    = threadIdx.x;
    const int mBlock = blockIdx.x * 64;

    // Load A tile 64x128 -> LDS (coalesced float4 reads, strided stores)
    for (int i = tid; i < 64 * 32; i += 256) {
        int row = i >> 5, c4 = i & 31;
        float4 v = ((const float4*)(A + (size_t)(mBlock + row) * CDIM))[c4];
        float* d = &As[row * 132 + c4 * 4];
        d[0] = v.x; d[1] = v.y; d[2] = v.z; d[3] = v.w;
    }

    const int wave = tid >> 5;
    const int lane = tid & 31;
    const int half = lane >> 4;      // selects K sub-pair per ISA A/B layout
    const int r    = lane & 15;
    const int mw   = (wave & 3) * 16;
    const int nh   = (wave >> 2) * 64;

    v8f zero = {0.f, 0.f, 0.f, 0.f, 0.f, 0.f, 0.f, 0.f};
    v8f acc[4] = {zero, zero, zero, zero};

    for (int kc = 0; kc < 4; ++kc) {           // K chunks of 32
        __syncthreads();
        // Transposed fill: Bst[col*36 + kl] = W[kc*32 + kl][col]
        for (int i = tid; i < 128 * 8; i += 256) {
            int col = i & 127, kg = i >> 7;    // kg = 0..7 (groups of 4 k)
            float4 v;
            v.x = W[(size_t)(kc * 32 + kg * 4 + 0) * CDIM + col];
            v.y = W[(size_t)(kc * 32 + kg * 4 + 1) * CDIM + col];
            v.z = W[(size_t)(kc * 32 + kg * 4 + 2) * CDIM + col];
            v.w = W[(size_t)(kc * 32 + kg * 4 + 3) * CDIM + col];
            *(float4*)&Bst[col * 36 + kg * 4] = v;
        }
        __syncthreads();

        #pragma unroll
        for (int kk = 0; kk < 8; ++kk) {       // 8 wmma K-steps per chunk
            const int k0 = kk * 4 + 2 * half;  // even -> 8B-aligned b64 loads
            v2f a = *(const v2f*)&As[(mw + r) * 132 + kc * 32 + k0];
            #pragma unroll
            for (int t = 0; t < 4; ++t) {
                const int col = nh + t * 16 + r;
                v2f b = *(const v2f*)&Bst[col * 36 + k0];
                acc[t] = __builtin_amdgcn_wmma_f32_16x16x4_f32(
                    false, a, false, b, (short)0, acc[t], false, false);
            }
        }
    }

    // D layout: VGPR i -> row i (lanes 0-15) / row i+8 (lanes 16-31)
    #pragma unroll
    for (int t = 0; t < 4; ++t) {
        #pragma unroll
        for (int i = 0; i < 8; ++i) {
            int row = mBlock + mw + i + half * 8;
            int col = nh + t * 16 + r;
            Out[(size_t)row * CDIM + col] = acc[t][i];
        }
    }
}

// ---------------------------------------------------------------------------
// Degree / normalization
// ---------------------------------------------------------------------------
__global__ void deg_init_k(float* __restrict__ deg, int N)
{
    int i = blockIdx.x * blockDim.x + threadIdx.x;
    if (i < N) deg[i] = 1.0f;            // self-loop
}

__global__ void deg_accum_k(float* __restrict__ deg,
                            const int* __restrict__ dl,
                            const int* __restrict__ grp,
                            const float* __restrict__ ew, int n)
{
    int e = blockIdx.x * blockDim.x + threadIdx.x;
    if (e >= EDGES) return;
    float w = ew ? ew[e] : 1.0f;
    if (w != 0.0f) atomicAdd(&deg[grp[e] * n + dl[e]], w);
}

__global__ void inv_sqrt_k(const float* __restrict__ deg,
                           float* __restrict__ dis, int N)
{
    int i = blockIdx.x * blockDim.x + threadIdx.x;
    if (i < N) dis[i] = rsqrtf(deg[i]);
}

// out[i,:] = xw[i,:] / deg[i] + bias  (self-loop term + bias, init for scatter)
__global__ void self_bias_k(const float* __restrict__ xw,
                            const float* __restrict__ dis,
                            const float* __restrict__ bias,
                            float* __restrict__ out, int N)
{
    int idx = blockIdx.x * blockDim.x + threadIdx.x;
    int node = idx >> 5, c4 = idx & 31;
    if (node >= N) return;
    float ds = dis[node];
    float invd = ds * ds;
    float4 t  = ((const float4*)(xw + (size_t)node * CDIM))[c4];
    float4 bv = ((const float4*)bias)[c4];
    float4 o;
    o.x = t.x * invd + bv.x;
    o.y = t.y * invd + bv.y;
    o.z = t.z * invd + bv.z;
    o.w = t.w * invd + bv.w;
    ((float4*)(out + (size_t)node * CDIM))[c4] = o;
}

// out[dst,:] += dis[src]*w*dis[dst] * xw[src,:]   (one wave32 per edge)
__global__ void edge_scatter_k(const float* __restrict__ xw,
                               const float* __restrict__ dis,
                               const int* __restrict__ sl,
                               const int* __restrict__ dl,
                               const int* __restrict__ grp,
                               const float* __restrict__ ew,
                               float* __restrict__ out, int n)
{
    int e = blockIdx.x * (blockDim.x >> 5) + (threadIdx.x >> 5);
    if (e >= EDGES) return;
    float w = ew ? ew[e] : 1.0f;
    if (w <= 0.0f) return;               // uniform across the wave
    int b = grp[e];
    int s = b * n + sl[e];
    int d = b * n + dl[e];
    float norm = dis[s] * w * dis[d];
    int lane = threadIdx.x & 31;
    float4 t = ((const float4*)(xw + (size_t)s * CDIM))[lane];
    float* dst = out + (size_t)d * CDIM + lane * 4;
    atomicAdd(dst + 0, norm * t.x);
    atomicAdd(dst + 1, norm * t.y);
    atomicAdd(dst + 2, norm * t.z);
    atomicAdd(dst + 3, norm * t.w);
}

// pn = p / ||p||
__global__ void p_norm_k(const float* __restrict__ p, float* __restrict__ pn)
{
    __shared__ float s[CDIM];
    int t = threadIdx.x;
    float v = p[t];
    s[t] = v * v;
    __syncthreads();
    for (int off = 64; off > 0; off >>= 1) {
        if (t < off) s[t] += s[t + off];
        __syncthreads();
    }
    pn[t] = v * rsqrtf(s[0]);
}

// in-place relu + score = tanh(h . pn)   (one wave32 per node)
__global__ void relu_score_k(float* __restrict__ h,
                             const float* __restrict__ pn,
                             float* __restrict__ score, int N)
{
    int node = blockIdx.x * (blockDim.x >> 5) + (threadIdx.x >> 5);
    if (node >= N) return;
    int lane = threadIdx.x & 31;
    float4* row = (float4*)(h + (size_t)node * CDIM);
    float4 t = row[lane];
    t.x = fmaxf(t.x, 0.f); t.y = fmaxf(t.y, 0.f);
    t.z = fmaxf(t.z, 0.f); t.w = fmaxf(t.w, 0.f);
    row[lane] = t;
    float4 pv = ((const float4*)pn)[lane];
    float d = t.x * pv.x + t.y * pv.y + t.z * pv.z + t.w * pv.w;
    for (int o = 16; o > 0; o >>= 1) d += __shfl_down(d, o, 32);
    if (lane == 0) score[node] = tanhf(d);
}

// per-graph top-k via in-LDS bitonic sort (descending); emits perm/vals/mapping
__launch_bounds__(1024)
__global__ void topk_sort_k(const float* __restrict__ score, int n, int k,
                            int* __restrict__ mapping,
                            int* __restrict__ perm,
                            float* __restrict__ vals)
{
    __shared__ float key[2048];
    __shared__ int   idx[2048];
    int b = blockIdx.x;
    for (int i = threadIdx.x; i < n; i += blockDim.x) {
        key[i] = score[b * n + i];
        idx[i] = i;
    }
    for (int size = 2; size <= n; size <<= 1) {
        for (int stride = size >> 1; stride > 0; stride >>= 1) {
            __syncthreads();
            for (int i = threadIdx.x; i < n; i += blockDim.x) {
                int j = i ^ stride;
                if (j > i) {
                    bool desc = ((i & size) == 0);
                    float ki = key[i], kj = key[j];
                    bool sw = desc ? (ki < kj) : (ki > kj);
                    if (sw) {
                        key[i] = kj; key[j] = ki;
                        int ti = idx[i]; idx[i] = idx[j]; idx[j] = ti;
                    }
                }
            }
        }
    }
    __syncthreads();
    for (int i = threadIdx.x; i < n; i += blockDim.x) mapping[b * n + i] = -1;
    __syncthreads();
    for (int j = threadIdx.x; j < k; j += blockDim.x) {
        int node = idx[j];
        mapping[b * n + node] = j;
        perm[b * k + j] = node;
        vals[b * k + j] = key[j];
    }
}

// hnext[b,j,:] = h[b, perm[b,j], :] * vals[b,j]   (one wave32 per output row)
__global__ void pool_gather_k(const float* __restrict__ h,
                              const int* __restrict__ perm,
                              const float* __restrict__ vals,
                              float* __restrict__ outp, int n, int k)
{
    int row = blockIdx.x * (blockDim.x >> 5) + (threadIdx.x >> 5);
    if (row >= BGRAPH * k) return;
    int lane = threadIdx.x & 31;
    int b = row / k;
    int node = perm[row];
    float v = vals[row];
    float4 t = ((const float4*)(h + ((size_t)b * n + node) * CDIM))[lane];
    t.x *= v; t.y *= v; t.z *= v; t.w *= v;
    ((float4*)(outp + (size_t)row * CDIM))[lane] = t;
}

// remap edges through pooling: filter_adj as weight mask
__global__ void edge_remap_k(const int* __restrict__ mapping,
                             const int* __restrict__ sl,
                             const int* __restrict__ dl,
                             const int* __restrict__ grp,
                             const float* __restrict__ ew, int n,
                             int* __restrict__ slo, int* __restrict__ dlo,
                             float* __restrict__ ewo)
{
    int e = blockIdx.x * blockDim.x + threadIdx.x;
    if (e >= EDGES) return;
    int b = grp[e];
    int ns = mapping[b * n + sl[e]];
    int nd = mapping[b * n + dl[e]];
    float w = ew ? ew[e] : 1.0f;
    bool valid = (w > 0.f) && (ns >= 0) && (nd >= 0);
    slo[e] = ns > 0 ? ns : 0;
    dlo[e] = nd > 0 ? nd : 0;
    ewo[e] = valid ? 1.0f : 0.0f;
}

// pooled[b,c] = sum_j h[b,j,c]
__global__ void pool_sum_k(const float* __restrict__ h,
                           float* __restrict__ pooled, int rows)
{
    int idx = blockIdx.x * blockDim.x + threadIdx.x;
    if (idx >= BGRAPH * CDIM) return;
    int b = idx >> 7, c = idx & 127;
    const float* p = h + (size_t)b * rows * CDIM + c;
    float s = 0.f;
    for (int j = 0; j < rows; ++j) s += p[(size_t)j * CDIM];
    pooled[idx] = s;
}

// out[b,o] = pooled[b,:] . Wm[:,o] + bm[o]
__global__ void mlp_out_k(const float* __restrict__ pooled,
                          const float* __restrict__ Wm,
                          const float* __restrict__ bm,
                          float* __restrict__ out)
{
    int idx = blockIdx.x * blockDim.x + threadIdx.x;
    if (idx >= BGRAPH * 16) return;
    int b = idx >> 4, o = idx & 15;
    float s = bm[o];
    const float* pr = pooled + b * CDIM;
    for (int c = 0; c < CDIM; ++c) s += pr[c] * Wm[c * 16 + o];
    out[idx] = s;
}

// ---------------------------------------------------------------------------
extern "C" void kernel_launch(void* const* d_in, const int* in_sizes, int n_in,
                              void* d_out, int out_size, void* d_ws, size_t ws_size,
                              hipStream_t stream)
{
    (void)in_sizes; (void)n_in; (void)out_size; (void)ws_size;

    const float* x     = (const float*)d_in[0];
    const int*   src_l = (const int*)d_in[1];
    const int*   dst_l = (const int*)d_in[2];
    const int*   grp   = (const int*)d_in[3];
    const float* Wl[3] = {(const float*)d_in[4], (const float*)d_in[7],  (const float*)d_in[10]};
    const float* bl[3] = {(const float*)d_in[5], (const float*)d_in[8],  (const float*)d_in[11]};
    const float* pl[3] = {(const float*)d_in[6], (const float*)d_in[9],  (const float*)d_in[12]};
    const float* Wm = (const float*)d_in[13];
    const float* bm = (const float*)d_in[14];
    float* out = (float*)d_out;

    // workspace carving (total ~211 MB)
    char* base = (char*)d_ws;
    size_t off = 0;
    auto carve = [&](size_t bytes) -> void* {
        void* p = base + off;
        off += (bytes + 255) & ~(size_t)255;
        return p;
    };
    float* buf0    = (float*)carve((size_t)131072 * CDIM * 4);  // xw
    float* buf1    = (float*)carve((size_t)131072 * CDIM * 4);  // conv out / relu
    float* buf2    = (float*)carve((size_t)65536  * CDIM * 4);  // pooled h
    float* deg     = (float*)carve(131072 * 4);
    float* dis     = (float*)carve(131072 * 4);
    float* score   = (float*)carve(131072 * 4);
    int*   mapping = (int*)  carve(131072 * 4);
    int*   perm    = (int*)  carve(65536 * 4);
    float* vals    = (float*)carve(65536 * 4);
    float* pn      = (float*)carve(512);
    int*   slA     = (int*)  carve((size_t)EDGES * 4);
    int*   dlA     = (int*)  carve((size_t)EDGES * 4);
    float* ewA     = (float*)carve((size_t)EDGES * 4);
    int*   slB     = (int*)  carve((size_t)EDGES * 4);
    int*   dlB     = (int*)  carve((size_t)EDGES * 4);
    float* ewB     = (float*)carve((size_t)EDGES * 4);
    float* pooled  = (float*)carve(BGRAPH * CDIM * 4);

    const float* hin = x;
    const int*   sl  = src_l;
    const int*   dl  = dst_l;
    const float* ew  = nullptr;
    int n = 2048;

    for (int l = 0; l < 3; ++l) {
        const int N = BGRAPH * n;
        const int k = n >> 1;

        gcn_gemm_wmma<<<N / 64, 256, 0, stream>>>(hin, Wl[l], buf0, N);
        deg_init_k   <<<N / 256, 256, 0, stream>>>(deg, N);
        deg_accum_k  <<<EDGES / 256, 256, 0, stream>>>(deg, dl, grp, ew, n);
        inv_sqrt_k   <<<N / 256, 256, 0, stream>>>(deg, dis, N);
        self_bias_k  <<<N / 8, 256, 0, stream>>>(buf0, dis, bl[l], buf1, N);
        edge_scatter_k<<<EDGES / 8, 256, 0, stream>>>(buf0, dis, sl, dl, grp, ew, buf1, n);
        p_norm_k     <<<1, 128, 0, stream>>>(pl[l], pn);
        relu_score_k <<<N / 8, 256, 0, stream>>>(buf1, pn, score, N);
        topk_sort_k  <<<BGRAPH, 1024, 0, stream>>>(score, n, k, mapping, perm, vals);
        pool_gather_k<<<(BGRAPH * k) / 8, 256, 0, stream>>>(buf1, perm, vals, buf2, n, k);

        if (l < 2) {
            int*   nsl = (l == 0) ? slA : slB;
            int*   ndl = (l == 0) ? dlA : dlB;
            float* new_ew = (l == 0) ? ewA : ewB;
            edge_remap_k<<<EDGES / 256, 256, 0, stream>>>(mapping, sl, dl, grp, ew, n,
                                                          nsl, ndl, new_ew);
            sl = nsl; dl = ndl; ew = new_ew;
        }
        hin = buf2;
        n = k;
    }

    // n == 256 here
    pool_sum_k<<<(BGRAPH * CDIM) / 256, 256, 0, stream>>>(buf2, pooled, n);
    mlp_out_k <<<(BGRAPH * 16 + 255) / 256, 256, 0, stream>>>(pooled, Wm, bm, out);
}